// IGANN_24670292148466
// MI455X (gfx1250) — compile-verified
//
#include <hip/hip_runtime.h>
#include <hip/hip_bf16.h>

typedef __attribute__((ext_vector_type(2))) float v2f;
typedef __attribute__((ext_vector_type(8))) float v8f;

#define F_FEAT 256
#define HID 16

// One wave = 16 batch rows. Per feature f:
//   A-frag (16x4 f32) = relu(x_m * W1[f,k] + b1[f,k]) built in-register (fc1 fused)
//   B-frag (4x16 f32) = W2[f, n, k]  (einsum 'bfh,fkh->bfk' => B[k][n] = W2[f,n,k])
//   C += A x B via 4x V_WMMA_F32_16X16X4_F32  (full-precision f32 matrix pipe)
//   epilogue: relu(C + b2[f,n]) * W3[f,n] accumulated per lane-column,
// then a 16-lane shfl_xor reduction over n at the end of the feature loop.
__global__ __launch_bounds__(256) void igann_wmma_kernel(
    const float* __restrict__ x,  const float* __restrict__ la,
    const float* __restrict__ bb, const float* __restrict__ W1,
    const float* __restrict__ b1, const float* __restrict__ W2,
    const float* __restrict__ b2, const float* __restrict__ W3,
    const float* __restrict__ b3, float* __restrict__ out, int batch)
{
    const int lane = threadIdx.x & 31;
    const int wave = threadIdx.x >> 5;
    const int half = lane >> 4;       // 0: lanes 0-15, 1: lanes 16-31
    const int lm   = lane & 15;       // A: row m; B/C: column n
    const int row0 = (blockIdx.x * 8 + wave) * 16;
    if (row0 >= batch) return;        // wave-uniform: EXEC stays all-1s inside

    const float* __restrict__ xrow = x + (size_t)(row0 + lm) * F_FEAT;

    float acc[8];
#pragma unroll
    for (int r = 0; r < 8; ++r) acc[r] = 0.f;
    float linacc = 0.f;               // sum_f x[m,f]*linear_a[f] (lane holds m=lm)
    float b3acc  = 0.f;               // sum_f b3[f] (wave-uniform)

    for (int f = 0; f < F_FEAT; ++f) {
        if ((f & 31) == 0) __builtin_prefetch(xrow + f + 32, 0, 1); // global_prefetch_b8

        const float xv = xrow[f];
        linacc = fmaf(xv, la[f], linacc);
        b3acc += b3[f];

        const float* __restrict__ w1f = W1 + f * HID;
        const float* __restrict__ b1f = b1 + f * HID;
        const float* __restrict__ w2f = W2 + f * HID * HID + lm * HID;

        v8f c = {0.f, 0.f, 0.f, 0.f, 0.f, 0.f, 0.f, 0.f};
#pragma unroll
        for (int j = 0; j < 4; ++j) {
            const int k0 = 4 * j + 2 * half;      // K index for VGPR0; VGPR1 = k0+1
            // fc1 fused into the A fragment (16x4 f32 layout)
            v2f w1 = *(const v2f*)(w1f + k0);
            v2f bi = *(const v2f*)(b1f + k0);
            v2f a;
            a.x = fmaxf(fmaf(xv, w1.x, bi.x), 0.f);
            a.y = fmaxf(fmaf(xv, w1.y, bi.y), 0.f);
            // B fragment: W2[f, n=lm, k0], k0+1 (contiguous, 8B aligned)
            v2f bfr = *(const v2f*)(w2f + k0);
            c = __builtin_amdgcn_wmma_f32_16x16x4_f32(
                    /*neg_a=*/false, a, /*neg_b=*/false, bfr,
                    /*c_mod=*/(short)0, c, /*reuse_a=*/false, /*reuse_b=*/false);
        }

        // fc3 epilogue on the C fragment: lane's column n=lm, rows r+8*half
        const float bb2 = b2[f * HID + lm];
        const float w3  = W3[f * HID + lm];
#pragma unroll
        for (int r = 0; r < 8; ++r) {
            float h2 = fmaxf(c[r] + bb2, 0.f);
            acc[r] = fmaf(h2, w3, acc[r]);
        }
    }

    const float biasv = bb[0];
#pragma unroll
    for (int r = 0; r < 8; ++r) {
        float v = acc[r];                 // reduce over n within each 16-lane half
        v += __shfl_xor(v, 1, 32);
        v += __shfl_xor(v, 2, 32);
        v += __shfl_xor(v, 4, 32);
        v += __shfl_xor(v, 8, 32);
        const int m = r + 8 * half;       // actual batch row within tile
        const float lin_m = __shfl(linacc, m, 32); // lane m holds row m's linear term
        if (lm == r) out[row0 + m] = v + lin_m + b3acc + biasv;
    }
}

extern "C" void kernel_launch(void* const* d_in, const int* in_sizes, int n_in,
                              void* d_out, int out_size, void* d_ws, size_t ws_size,
                              hipStream_t stream) {
    const float* x  = (const float*)d_in[0];
    const float* la = (const float*)d_in[1];
    const float* bb = (const float*)d_in[2];
    const float* W1 = (const float*)d_in[3];
    const float* b1 = (const float*)d_in[4];
    const float* W2 = (const float*)d_in[5];
    const float* b2 = (const float*)d_in[6];
    const float* W3 = (const float*)d_in[7];
    const float* b3 = (const float*)d_in[8];
    float* out = (float*)d_out;

    const int batch = in_sizes[0] / F_FEAT;       // 32768
    const int rowsPerBlock = 128;                 // 8 waves * 16 rows
    const int grid = (batch + rowsPerBlock - 1) / rowsPerBlock;
    igann_wmma_kernel<<<grid, 256, 0, stream>>>(x, la, bb, W1, b1, W2, b2, W3,
                                                b3, out, batch);
}